// HeteroGCN_4114578669780
// MI455X (gfx1250) — compile-verified
//
#include <hip/hip_runtime.h>

typedef __attribute__((ext_vector_type(2))) float v2f;
typedef __attribute__((ext_vector_type(8))) float v8f;

#define CDIM 128
#define NUSER 100000
#define NITEM 50000
#define NEDGE 500000
#define NLAYER 2
#define BSEED 1024
#define APITCH 258   // 16x256 A tile, padded (258 % 64 == 2 -> conflict-free row reads)
#define OPITCH 132   // 16x128 D tile, padded
#define WPK_PER 32768   // floats per packed (layer,edge-type) weight block: 256*128

// ---------------------------------------------------------------- zero
__global__ void hgcn_zero(float* __restrict__ p, long n) {
  long idx = ((long)blockIdx.x * blockDim.x + threadIdx.x) * 4;
  if (idx + 3 < n) {
    *(float4*)(p + idx) = make_float4(0.f, 0.f, 0.f, 0.f);
  } else {
    for (long j = idx; j < n; ++j) p[j] = 0.f;
  }
}

// ---------------------------------------------------------------- weight packing
// For each (l,et): K = [w_rel ; w_root] (256 x 128). Store row pairs interleaved:
//   dst[le][p*256 + col*2 + (k&1)] = K[k][col],  p = k/2
// so one lane's B operand {K[k0][col], K[k0+1][col]} (k0 even) is one contiguous b64.
__global__ void hgcn_pack_w(const float* __restrict__ w_rel,
                            const float* __restrict__ w_root,
                            float* __restrict__ dst) {
  int gid = blockIdx.x * blockDim.x + threadIdx.x;   // over 4*256*128
  if (gid >= 4 * 256 * CDIM) return;
  int le  = gid >> 15;            // (l*2+et), 0..3
  int rem = gid & 32767;
  int k   = rem >> 7;             // 0..255
  int col = rem & 127;
  float v = (k < CDIM) ? w_rel[(long)le * CDIM * CDIM + (long)k * CDIM + col]
                       : w_root[(long)le * CDIM * CDIM + (long)(k - CDIM) * CDIM + col];
  dst[(long)le * WPK_PER + (long)(k >> 1) * 256 + col * 2 + (k & 1)] = v;
}

// ---------------------------------------------------------------- scatter-add (segment_sum)
__global__ void hgcn_scatter(const float* __restrict__ x,
                             const int* __restrict__ src,
                             const int* __restrict__ dst,
                             float* __restrict__ agg, int nE) {
  int gid = blockIdx.x * blockDim.x + threadIdx.x;
  int e = gid >> 5;
  if (e >= nE) return;
  int g = (gid & 31) * 4;
  int s = src[e], d = dst[e];
  float4 v = *(const float4*)(x + (long)s * CDIM + g);
  float* ap = agg + (long)d * CDIM + g;
  atomicAdd(ap + 0, v.x);
  atomicAdd(ap + 1, v.y);
  atomicAdd(ap + 2, v.z);
  atomicAdd(ap + 3, v.w);
}

// ---------------------------------------------------------------- fused GEMM + bias + LN + ReLU
// out[node] = relu(LN([agg[node] | x[node]] @ Kpacked + bias))
// Block = 256 threads (8 waves), 16 node rows; wave w -> output cols [16w,16w+16).
__global__ __launch_bounds__(256)
void hgcn_gemm_ln(const float* agg,                  // [N,128]  (also written as out)
                  const float* __restrict__ x,       // [N,128]
                  const float* __restrict__ wpk,     // packed 256x128 weights (row pairs)
                  const float* __restrict__ bias,    // [128]
                  const float* __restrict__ gam,     // [128]
                  const float* __restrict__ beta,    // [128]
                  float* out,                        // [N,128] (aliases agg)
                  int N) {
  __shared__ float sA[16 * APITCH];
  __shared__ float sO[16 * OPITCH];
  const int tid = threadIdx.x;
  const int block_row = blockIdx.x * 16;

  // Stage A = [agg_rows | x_rows] (16 x 256) into LDS, coalesced.
  for (int i = 0; i < 16; ++i) {
    int flat = i * 256 + tid;
    int r = flat >> 8;
    int col = flat & 255;
    int node = block_row + r;
    float v = 0.f;
    if (node < N)
      v = (col < CDIM) ? agg[(long)node * CDIM + col]
                       : x[(long)node * CDIM + (col - CDIM)];
    sA[r * APITCH + col] = v;
  }
  __syncthreads();

  const int wave = tid >> 5;
  const int lane = tid & 31;
  const int half = lane >> 4;      // lanes 16-31 hold K+2 / M+8 per ISA layouts
  const int mr = lane & 15;        // A row
  const int nc = lane & 15;        // B/D column within tile
  const int n0 = wave * 16;

  // Per-lane bases: A at sA[mr][2*half + kb]; B pair index p = kb/2 + half.
  const float* aBase = &sA[mr * APITCH + 2 * half];
  const float* bBase = wpk + (long)half * 256 + (n0 + nc) * 2;

  v8f c = {};
#pragma unroll 8
  for (int kb = 0; kb < 256; kb += 4) {
    v2f a = *(const v2f*)(aBase + kb);
    v2f b = *(const v2f*)(bBase + (kb >> 1) * 256);
    c = __builtin_amdgcn_wmma_f32_16x16x4_f32(
        false, a, false, b, (short)0, c, false, false);
  }

  // bias (one output column per lane)
  float bb = bias[n0 + nc];
#pragma unroll
  for (int v = 0; v < 8; ++v) c[v] += bb;

  // D tile -> LDS: lane (half, nc), VGPR v -> row v + 8*half
#pragma unroll
  for (int v = 0; v < 8; ++v)
    sO[(v + 8 * half) * OPITCH + n0 + nc] = c[v];
  __syncthreads();

  // LayerNorm + ReLU: wave w handles rows 2w, 2w+1
  for (int rr = 0; rr < 2; ++rr) {
    int r = wave * 2 + rr;
    int node = block_row + r;
    float vals[4];
    float s = 0.f, sq = 0.f;
#pragma unroll
    for (int j = 0; j < 4; ++j) {
      float v = sO[r * OPITCH + lane + j * 32];
      vals[j] = v;
      s += v;
      sq += v * v;
    }
#pragma unroll
    for (int off = 16; off > 0; off >>= 1) {
      s += __shfl_xor(s, off, 32);
      sq += __shfl_xor(sq, off, 32);
    }
    float mean = s * (1.f / 128.f);
    float var = sq * (1.f / 128.f) - mean * mean;
    float rstd = rsqrtf(var + 1e-5f);
    if (node < N) {
#pragma unroll
      for (int j = 0; j < 4; ++j) {
        int col = lane + j * 32;
        float v = (vals[j] - mean) * rstd * gam[col] + beta[col];
        out[(long)node * CDIM + col] = fmaxf(v, 0.f);
      }
    }
  }
}

// ---------------------------------------------------------------- head: xu[:B] @ lin_w + lin_b
__global__ void hgcn_head(const float* __restrict__ xu,
                          const float* __restrict__ lw,
                          const float* __restrict__ lb,
                          float* __restrict__ out) {
  int wave = threadIdx.x >> 5;
  int lane = threadIdx.x & 31;
  int row = blockIdx.x * 8 + wave;
  if (row >= BSEED) return;
  float s = 0.f;
#pragma unroll
  for (int j = 0; j < 4; ++j) {
    int col = lane + j * 32;
    s += xu[(long)row * CDIM + col] * lw[col];
  }
#pragma unroll
  for (int off = 16; off > 0; off >>= 1) s += __shfl_xor(s, off, 32);
  if (lane == 0) out[row] = s + lb[0];
}

// ----------------------------------------------------------------
extern "C" void kernel_launch(void* const* d_in, const int* in_sizes, int n_in,
                              void* d_out, int out_size, void* d_ws, size_t ws_size,
                              hipStream_t stream) {
  const float* x_user  = (const float*)d_in[0];
  const float* x_item  = (const float*)d_in[1];
  const int*   eui_src = (const int*)d_in[2];
  const int*   eui_dst = (const int*)d_in[3];
  const int*   eiu_src = (const int*)d_in[4];
  const int*   eiu_dst = (const int*)d_in[5];
  const float* w_rel   = (const float*)d_in[6];   // [L,2,C,C]
  const float* w_root  = (const float*)d_in[7];   // [L,2,C,C]
  const float* bvec    = (const float*)d_in[8];   // [L,2,C]
  const float* ln_g    = (const float*)d_in[9];   // [L,2,C] (0=user,1=item)
  const float* ln_b    = (const float*)d_in[10];
  const float* lin_w   = (const float*)d_in[11];
  const float* lin_b   = (const float*)d_in[12];
  float* out = (float*)d_out;

  float* ws = (float*)d_ws;
  const long nU = (long)NUSER * CDIM;   // 12.8M floats
  const long nI = (long)NITEM * CDIM;   // 6.4M floats
  float* xu_buf[2] = { ws, ws + nU };
  float* xi_buf[2] = { ws + 2 * nU, ws + 2 * nU + nI };
  float* wpk = ws + 2 * nU + 2 * nI;    // 4 * 32768 floats of packed weights

  // Pack all 4 (layer, edge-type) weight blocks once.
  hgcn_pack_w<<<(4 * 256 * CDIM + 255) / 256, 256, 0, stream>>>(w_rel, w_root, wpk);

  const float* cur_xu = x_user;
  const float* cur_xi = x_item;
  const int sblocks = (NEDGE * 32 + 255) / 256;

  for (int l = 0; l < NLAYER; ++l) {
    float* agg_u = xu_buf[l];
    float* agg_i = xi_buf[l];
    hgcn_zero<<<(int)((nU / 4 + 255) / 256), 256, 0, stream>>>(agg_u, nU);
    hgcn_zero<<<(int)((nI / 4 + 255) / 256), 256, 0, stream>>>(agg_i, nI);

    // agg_i = segsum(xu[eui_src] -> eui_dst),  agg_u = segsum(xi[eiu_src] -> eiu_dst)
    hgcn_scatter<<<sblocks, 256, 0, stream>>>(cur_xu, eui_src, eui_dst, agg_i, NEDGE);
    hgcn_scatter<<<sblocks, 256, 0, stream>>>(cur_xi, eiu_src, eiu_dst, agg_u, NEDGE);

    long c0 = ((long)l * 2 + 0) * CDIM;   // param row (l,0)
    long c1 = ((long)l * 2 + 1) * CDIM;   // param row (l,1)

    // items: weights/bias index (l,0), LN index (l,1)
    hgcn_gemm_ln<<<NITEM / 16, 256, 0, stream>>>(agg_i, cur_xi,
        wpk + (long)(l * 2 + 0) * WPK_PER, bvec + c0, ln_g + c1, ln_b + c1, agg_i, NITEM);
    // users: weights/bias index (l,1), LN index (l,0)
    hgcn_gemm_ln<<<NUSER / 16, 256, 0, stream>>>(agg_u, cur_xu,
        wpk + (long)(l * 2 + 1) * WPK_PER, bvec + c1, ln_g + c0, ln_b + c0, agg_u, NUSER);

    cur_xu = agg_u;
    cur_xi = agg_i;
  }

  hgcn_head<<<BSEED / 8, 256, 0, stream>>>(cur_xu, lin_w, lin_b, out);
}